// GCNEncoder_44358422233325
// MI455X (gfx1250) — compile-verified
//
#include <hip/hip_runtime.h>

typedef __attribute__((ext_vector_type(2))) float v2f;
typedef __attribute__((ext_vector_type(8))) float v8f;

// ---------------------------------------------------------------------------
// Degree / normalization kernels
// ---------------------------------------------------------------------------
__global__ void deg_init_kernel(float* __restrict__ dinv, int n) {
    int i = blockIdx.x * blockDim.x + threadIdx.x;
    if (i < n) dinv[i] = 1.0f;          // self-loop contributes 1 to deg
}

__global__ void deg_count_kernel(const long long* __restrict__ ei,
                                 float* __restrict__ dinv, long long E) {
    long long e = (long long)blockIdx.x * blockDim.x + threadIdx.x;
    if (e < E) {
        int d = (int)ei[E + e];         // dst row of edge_index
        atomicAdd(&dinv[d], 1.0f);
    }
}

__global__ void deg_finalize_kernel(float* __restrict__ dinv, int n) {
    int i = blockIdx.x * blockDim.x + threadIdx.x;
    if (i < n) dinv[i] = rsqrtf(dinv[i]);   // deg >= 1 always (self loop)
}

// ---------------------------------------------------------------------------
// Fused GEMM (WMMA f32 16x16x4) + self-loop epilogue.
//   H   = act(A) @ W                    (act = relu(x + b_in) if RELU_IN)
//   AGG = dinv[row]^2 * H              (self-loop init of the aggregator)
// One wave computes 16 rows x NOUT cols; 8 waves/WG -> 128 rows per block.
//
// W is staged into LDS in B-fragment order: for k-pair row kp, column col,
// the pair [W[2kp][col], W[2kp+1][col]] is adjacent -> one ds_load_b64 per
// B fragment, no register shuffles. Bit 5 of the within-row dword offset is
// XOR-swizzled by (kp&1) so the two half-waves (which read adjacent kp rows)
// hit disjoint bank halves. Since k steps by 4, kp&1 == halfsel is a
// loop-invariant lane constant, folded into two base pointers.
// ---------------------------------------------------------------------------
template <int KDIM, int NOUT, bool RELU_IN>
__global__ void __launch_bounds__(256)
gcn_gemm_wmma_kernel(const float* __restrict__ A,      // [nrows, KDIM]
                     const float* __restrict__ W,      // [KDIM, NOUT]
                     const float* __restrict__ bias_in,// [KDIM] (RELU_IN only)
                     const float* __restrict__ dinv,   // [nrows]
                     float* __restrict__ H,            // [nrows, NOUT]
                     float* __restrict__ AGG,          // [nrows, NOUT]
                     int nrows) {
    constexpr int NT = NOUT / 16;                 // 16-wide column tiles
    constexpr int RP = 2 * NOUT;                  // dwords per kp-row
    __shared__ __align__(16) float wlds[(KDIM / 2) * RP];

    // Stage W pair-interleaved + bank-swizzled:
    //   wlds[kp*RP + ((col*2 + parity) ^ ((kp&1)*32))] = W[2kp+parity][col]
    for (int i = threadIdx.x; i < KDIM * NOUT; i += blockDim.x) {
        const int k      = i / NOUT;
        const int col    = i - k * NOUT;
        const int kp     = k >> 1;
        const int parity = k & 1;
        wlds[kp * RP + (((col << 1) + parity) ^ ((kp & 1) << 5))] = W[i];
    }
    __syncthreads();

    const int lane    = threadIdx.x & 31;
    const int wave    = threadIdx.x >> 5;
    const int rowBase = (blockIdx.x * 8 + wave) * 16;
    const int halfsel = lane >> 4;                // 0: lanes 0-15, 1: 16-31
    const int l15     = lane & 15;

    const v8f vzero = {};
    v8f acc[NT];
#pragma unroll
    for (int t = 0; t < NT; ++t) acc[t] = vzero;

    // A fragment: lane holds row (rowBase + l15), k = kbase + halfsel*2 {+0,+1}
    const int  arow    = rowBase + l15;
    const bool arow_ok = arow < nrows;
    const size_t abase = (size_t)(arow_ok ? arow : 0) * KDIM;

    for (int k = 0; k < KDIM; k += 4) {
        const int kk = k + halfsel * 2;
        float ax = A[abase + kk];
        float ay = A[abase + kk + 1];
        if constexpr (RELU_IN) {
            ax = fmaxf(ax + bias_in[kk], 0.0f);
            ay = fmaxf(ay + bias_in[kk + 1], 0.0f);
        }
        if (!arow_ok) { ax = 0.0f; ay = 0.0f; }   // zero-pad OOB rows (EXEC stays full)
        v2f a; a.x = ax; a.y = ay;

        // kp&1 == halfsel (k>>1 is even), so the swizzle constant is lane-
        // invariant across the loop; fold it into even/odd-tile base pointers.
        const int kp = (k >> 1) + halfsel;
        const float* bbase = &wlds[kp * RP + (l15 << 1)];
        const float* bE = bbase + (halfsel << 5);   // even t: offset +h*32
        const float* bO = bbase - (halfsel << 5);   // odd  t: offset -h*32

#pragma unroll
        for (int t = 0; t < NT; ++t) {
            const float* bp = ((t & 1) ? bO : bE) + t * 32;
            const v2f b = *(const v2f*)bp;          // single ds_load_b64
            acc[t] = __builtin_amdgcn_wmma_f32_16x16x4_f32(
                false, a, false, b, (short)0, acc[t], false, false);
        }
    }

    // Epilogue: D layout -> VGPR v holds row (rowBase + v + 8*halfsel), col l15
#pragma unroll
    for (int v = 0; v < 8; ++v) {
        const int row = rowBase + v + 8 * halfsel;
        if (row < nrows) {
            const float di  = dinv[row];
            const float di2 = di * di;
            const size_t rb = (size_t)row * NOUT;
#pragma unroll
            for (int t = 0; t < NT; ++t) {
                const int   col = t * 16 + l15;
                const float val = acc[t][v];
                H[rb + col]   = val;
                AGG[rb + col] = di2 * val;
            }
        }
    }
}

// ---------------------------------------------------------------------------
// Edge scatter: AGG[dst] += dinv[src]*dinv[dst] * H[src].
// NOUT/4 lanes per edge, float4 per lane. src/dst loads are uniform across an
// edge's lanes; gathers are coalesced 128B lines and L2-hot (H < 192MB L2).
// ---------------------------------------------------------------------------
template <int NOUT>
__global__ void gcn_scatter_kernel(const long long* __restrict__ ei, // [2,E]
                                   const float* __restrict__ H,
                                   const float* __restrict__ dinv,
                                   float* __restrict__ AGG,
                                   long long E) {
    constexpr int LPE = NOUT / 4;                       // lanes per edge
    const long long tid = (long long)blockIdx.x * blockDim.x + threadIdx.x;
    const long long e   = tid / LPE;
    const int       sub = (int)(tid % LPE);
    if (e >= E) return;

    const int s = (int)ei[e];
    const int d = (int)ei[E + e];
    const float w = dinv[s] * dinv[d];

    const float4 hv = *(const float4*)(H + (size_t)s * NOUT + sub * 4);
    float* dst = AGG + (size_t)d * NOUT + sub * 4;
    atomicAdd(dst + 0, w * hv.x);
    atomicAdd(dst + 1, w * hv.y);
    atomicAdd(dst + 2, w * hv.z);
    atomicAdd(dst + 3, w * hv.w);
}

// ---------------------------------------------------------------------------
// Final bias add: out = agg2 + b2   (N x 64)
// ---------------------------------------------------------------------------
__global__ void bias_out_kernel(const float* __restrict__ agg,
                                const float* __restrict__ b,
                                float* __restrict__ out, int n) {
    int i = blockIdx.x * blockDim.x + threadIdx.x;
    if (i < n) out[i] = agg[i] + b[i & 63];
}

// ---------------------------------------------------------------------------
// Host launcher
// ---------------------------------------------------------------------------
extern "C" void kernel_launch(void* const* d_in, const int* in_sizes, int n_in,
                              void* d_out, int out_size, void* d_ws, size_t ws_size,
                              hipStream_t stream) {
    (void)n_in; (void)out_size; (void)ws_size;
    const float*     x  = (const float*)d_in[0];          // [N,128]
    const long long* ei = (const long long*)d_in[1];      // [2,E] int64
    const float*     W1 = (const float*)d_in[2];          // [128,128]
    const float*     b1 = (const float*)d_in[3];          // [128]
    const float*     W2 = (const float*)d_in[4];          // [128,64]
    const float*     b2 = (const float*)d_in[5];          // [64]
    float*           out = (float*)d_out;                 // [N,64]

    const int       N = in_sizes[0] / 128;
    const long long E = (long long)in_sizes[1] / 2;

    // Workspace layout (floats): dinv[N] | buf1[128N] | agg1[128N]
    //   buf1 = h1 during layer 1; reused as h2[64N] + agg2[64N] in layer 2.
    float* ws   = (float*)d_ws;
    float* dinv = ws;
    float* h1   = ws + N;
    float* agg1 = ws + N + (size_t)N * 128;
    float* h2   = h1;                      // h1 dead after scatter1
    float* agg2 = h1 + (size_t)N * 64;

    const int T = 256;
    // 1. normalization: deg -> dinv
    deg_init_kernel<<<(N + T - 1) / T, T, 0, stream>>>(dinv, N);
    deg_count_kernel<<<(int)((E + T - 1) / T), T, 0, stream>>>(ei, dinv, E);
    deg_finalize_kernel<<<(N + T - 1) / T, T, 0, stream>>>(dinv, N);

    // 2. layer 1: h1 = x@W1 ; agg1 = dinv^2*h1 ; scatter edges
    gcn_gemm_wmma_kernel<128, 128, false>
        <<<(N + 127) / 128, T, 0, stream>>>(x, W1, nullptr, dinv, h1, agg1, N);
    {
        const long long threads = E * 32;               // 32 lanes/edge (F=128)
        gcn_scatter_kernel<128>
            <<<(int)((threads + T - 1) / T), T, 0, stream>>>(ei, h1, dinv, agg1, E);
    }

    // 3. layer 2: A-load applies relu(agg1 + b1); h2 = A@W2 ; agg2 = dinv^2*h2
    gcn_gemm_wmma_kernel<128, 64, true>
        <<<(N + 127) / 128, T, 0, stream>>>(agg1, W2, b1, dinv, h2, agg2, N);
    {
        const long long threads = E * 16;               // 16 lanes/edge (F=64)
        gcn_scatter_kernel<64>
            <<<(int)((threads + T - 1) / T), T, 0, stream>>>(ei, h2, dinv, agg2, E);
    }

    // 4. out = agg2 + b2
    bias_out_kernel<<<(N * 64 + T - 1) / T, T, 0, stream>>>(agg2, b2, out, N * 64);
}